// MF_MPC_57148834841202
// MI455X (gfx1250) — compile-verified
//
#include <hip/hip_runtime.h>
#include <hip/hip_bf16.h>

// MF-MPC forward on MI455X (gfx1250, wave32).
// Phase A: row_sum[t] = sum_d Mr[flat_items[t]][d]   (gather-bound, L2-resident table)
// Phase B: per-segment masked sum / sqrt(count) via binary search on sorted seg_ids
// Phase C: batched 64-dim dot via chained V_WMMA_F32_16X16X4_F32 (diagonal extract)

typedef __attribute__((ext_vector_type(2))) float v2f;
typedef __attribute__((ext_vector_type(8))) float v8f;

#define R_LEVELS 5
#define DIM 64

// ---------------- Phase A: per-element row sums ----------------
// 8 lanes per element: lane j loads Mr[row][4j..4j+3] and Mr[row][32+4j..32+4j+3]
// -> two contiguous 128B bursts per 8-lane group, then shfl_xor tree reduce.
__global__ void mfmpc_rowsum(const int* __restrict__ flat_items,
                             const float* __restrict__ Mr,
                             float* __restrict__ rs, int T) {
  const int tid  = blockIdx.x * blockDim.x + threadIdx.x;
  const int wave = tid >> 5;
  const int lane = tid & 31;
  const int grp  = lane >> 3;   // element-group within wave (0..3)
  const int j    = lane & 7;    // lane within group
  const int t    = wave * 4 + grp;
  if (t >= T) return;

  const int item = flat_items[t];
  const float4* row = (const float4*)(Mr + (size_t)item * DIM);
  float4 x = row[j];       // dims [4j, 4j+4)
  float4 y = row[j + 8];   // dims [32+4j, 32+4j+4)
  float s = (x.x + x.y) + (x.z + x.w) + (y.x + y.y) + (y.z + y.w);
  // reduce across the aligned 8-lane group
  s += __shfl_xor(s, 1);
  s += __shfl_xor(s, 2);
  s += __shfl_xor(s, 4);
  if (j == 0) rs[t] = s;
}

// ---------------- Phase B: segment reduction ----------------
__device__ __forceinline__ int lower_bound_i32(const int* __restrict__ a, int n, int v) {
  int lo = 0, hi = n;
  while (lo < hi) {
    int mid = (lo + hi) >> 1;
    if (a[mid] < v) lo = mid + 1; else hi = mid;
  }
  return lo;
}

__global__ void mfmpc_segreduce(const int* __restrict__ seg_ids,
                                const int* __restrict__ flat_items,
                                const int* __restrict__ item_idx,
                                const float* __restrict__ rs,
                                float* __restrict__ per_seg,
                                int T, int nseg) {
  const int s = blockIdx.x * blockDim.x + threadIdx.x;
  if (s >= nseg) return;

  const int start = lower_bound_i32(seg_ids, T, s);
  const int end   = lower_bound_i32(seg_ids, T, s + 1);
  const int target = item_idx[s / R_LEVELS];

  float sum = 0.0f;
  int cnt = 0;
  for (int t = start; t < end; ++t) {
    const int fi = flat_items[t];
    if (fi != target) { sum += rs[t]; ++cnt; }
  }
  per_seg[s] = (cnt > 0) ? (sum / sqrtf((float)cnt)) : 0.0f;
}

// ---------------- Phase C: WMMA batched dot + bias + clip ----------------
// One wave handles 16 samples. A = user_vec tile (16x4 per step), B = item_vec
// tile (4x16 per step), chained over 16 K-chunks of DIM=64. Diagonal of the
// 16x16 f32 accumulator = the 16 dot products.
__global__ void mfmpc_predict(const int* __restrict__ user_idx,
                              const int* __restrict__ item_idx,
                              const float* __restrict__ user_emb,
                              const float* __restrict__ item_emb,
                              const float* __restrict__ user_bias,
                              const float* __restrict__ item_bias,
                              const float* __restrict__ g_avg,
                              const float* __restrict__ per_seg,
                              float* __restrict__ out, int Bn) {
  const int tid  = blockIdx.x * blockDim.x + threadIdx.x;
  const int wave = tid >> 5;
  const int lane = tid & 31;
  const int base = wave * 16;
  if (base >= Bn) return;          // wave-uniform: EXEC stays all-ones for WMMA

  const int m    = lane & 15;      // sample within the 16-wide tile
  const int half = lane >> 4;      // 0 -> K{0,1}, 1 -> K{2,3}
  const int b    = base + m;

  const int u  = user_idx[b];
  const int it = item_idx[b];

  // uu_mpc[b] = sum of the 5 rating-level segment values (scalar broadcast)
  const float* ps = per_seg + (size_t)b * R_LEVELS;
  const float uu = ((ps[0] + ps[1]) + (ps[2] + ps[3])) + ps[4];

  const float* urow = user_emb + (size_t)u * DIM;
  const float* irow = item_emb + (size_t)it * DIM;

  v8f c = {};
  #pragma unroll
  for (int ch = 0; ch < 16; ++ch) {
    const int d0 = ch * 4 + half * 2;
    v2f a, bb;
    a[0]  = urow[d0]     + uu;     // A[m][d0]   (lane layout per ISA 16x4 f32)
    a[1]  = urow[d0 + 1] + uu;     // A[m][d0+1]
    bb[0] = irow[d0];              // B[d0][m]   (column-striped by lane)
    bb[1] = irow[d0 + 1];
    c = __builtin_amdgcn_wmma_f32_16x16x4_f32(
        /*neg_a=*/false, a, /*neg_b=*/false, bb,
        /*c_mod=*/(short)0, c, /*reuse_a=*/false, /*reuse_b=*/false);
  }

  // Diagonal extraction: D[i][i] lives in lane i, VGPR i (i<8) and
  // lane 16+i, VGPR i-8 (i>=8). Each such lane already owns sample base+(lane&15).
  float diag = 0.0f;
  bool has = false;
  if (lane < 8)        { diag = c[lane];      has = true; }
  else if (lane >= 24) { diag = c[lane - 24]; has = true; }

  if (has) {
    float r = diag + user_bias[u] + item_bias[it] + g_avg[0];
    r = fminf(fmaxf(r, 1.0f), 5.0f);
    out[b] = r;
  }
}

extern "C" void kernel_launch(void* const* d_in, const int* in_sizes, int n_in,
                              void* d_out, int out_size, void* d_ws, size_t ws_size,
                              hipStream_t stream) {
  const int*   user_idx   = (const int*)  d_in[0];
  const int*   item_idx   = (const int*)  d_in[1];
  const int*   flat_items = (const int*)  d_in[2];
  const int*   seg_ids    = (const int*)  d_in[3];
  const float* user_emb   = (const float*)d_in[4];
  const float* item_emb   = (const float*)d_in[5];
  const float* Mr         = (const float*)d_in[6];
  const float* user_bias  = (const float*)d_in[7];
  const float* item_bias  = (const float*)d_in[8];
  const float* g_avg      = (const float*)d_in[9];
  float* out = (float*)d_out;

  const int T  = in_sizes[2];          // 1,638,400
  const int Bn = in_sizes[0];          // 16,384
  const int nseg = Bn * R_LEVELS;      // 81,920

  // workspace layout: rs[T] floats, then per_seg[nseg] floats (~6.9 MB total)
  float* rs      = (float*)d_ws;
  float* per_seg = rs + T;

  // Phase A: 8 lanes per element -> T*8 threads
  {
    const long long threads = (long long)T * 8;
    const int blocks = (int)((threads + 255) / 256);
    mfmpc_rowsum<<<blocks, 256, 0, stream>>>(flat_items, Mr, rs, T);
  }
  // Phase B: one thread per segment
  {
    const int blocks = (nseg + 255) / 256;
    mfmpc_segreduce<<<blocks, 256, 0, stream>>>(seg_ids, flat_items, item_idx,
                                                rs, per_seg, T, nseg);
  }
  // Phase C: one wave per 16 samples
  {
    const int waves = (Bn + 15) / 16;
    const int blocks = (waves * 32 + 255) / 256;
    mfmpc_predict<<<blocks, 256, 0, stream>>>(user_idx, item_idx, user_emb,
                                              item_emb, user_bias, item_bias,
                                              g_avg, per_seg, out, Bn);
  }
}